// GNN_88622355185755
// MI455X (gfx1250) — compile-verified
//
#include <hip/hip_runtime.h>
#include <hip/hip_bf16.h>

#define EPSV 1e-5f
#define BDIM 256
#define TDIM 300
#define NKP  17
#define NEDGE 19

typedef __attribute__((ext_vector_type(2))) float v2f;
typedef __attribute__((ext_vector_type(8))) float v8f;

// ---------------- wave32 reduction ----------------
__device__ __forceinline__ float wsum32(float v) {
  for (int off = 16; off > 0; off >>= 1) v += __shfl_xor(v, off, 32);
  return v;
}

// ---------------- Stage 1: per-frame keypoint GCN ----------------
// One wave per graph (8 graphs per 256-thread block). Lane = node (lanes 0..16 active).
template<int CIN, int COUT>
__device__ __forceinline__ void gcn_layer(float* xw_g,          // [17*16] this wave's LDS tile
                                          const float* dinv_g,  // [17]
                                          const int* src_g, const int* dst_g,
                                          int lane,
                                          const float* hin,     // [CIN] per-lane (node)
                                          const float* __restrict__ W,   // [CIN,COUT]
                                          const float* __restrict__ bias,// [COUT]
                                          float* hout)          // [COUT]
{
  if (lane < NKP) {
    for (int c = 0; c < COUT; ++c) {
      float s = 0.f;
      for (int k = 0; k < CIN; ++k) s += hin[k] * W[k * COUT + c];
      xw_g[lane * 16 + c] = s;
    }
  }
  __syncthreads();
  float acc[COUT];
  for (int c = 0; c < COUT; ++c) acc[c] = 0.f;
  if (lane < NKP) {
    float dn = dinv_g[lane];
    float nn = dn * dn;                      // self-loop norm
    for (int c = 0; c < COUT; ++c) acc[c] = nn * xw_g[lane * 16 + c];
    for (int e = 0; e < NEDGE; ++e) {
      int s = src_g[e];
      int d = dst_g[e];
      if (d == lane) {
        float w = dinv_g[s] * dn;
        for (int c = 0; c < COUT; ++c) acc[c] += w * xw_g[s * 16 + c];
      }
    }
    for (int c = 0; c < COUT; ++c) acc[c] += bias[c];
  }
  __syncthreads();                           // xw reads done before next layer rewrites
  // graph BN (weight=1, bias=0, population var over 17 nodes) + ReLU
  for (int c = 0; c < COUT; ++c) {
    float sm = wsum32(acc[c]);
    float sq = wsum32(acc[c] * acc[c]);
    float m  = sm * (1.f / 17.f);
    float vr = sq * (1.f / 17.f) - m * m;
    float o  = (acc[c] - m) * rsqrtf(vr + EPSV);
    hout[c]  = fmaxf(o, 0.f);
  }
}

__global__ void gcn3_kernel(const float* __restrict__ x,   // [B*T, 17, 2]
                            const int*   __restrict__ ei,  // [B*T, 2, 19]
                            const float* __restrict__ W1, const float* __restrict__ b1,
                            const float* __restrict__ W2, const float* __restrict__ b2,
                            const float* __restrict__ W3, const float* __restrict__ b3,
                            float* __restrict__ x_time)    // [B*T, 68]
{
  __shared__ float s_xw[8][NKP * 16];
  __shared__ float s_dinv[8][NKP];
  __shared__ int   s_src[8][NEDGE];
  __shared__ int   s_dst[8][NEDGE];

  const int wid  = threadIdx.x >> 5;
  const int lane = threadIdx.x & 31;
  const long g   = (long)blockIdx.x * 8 + wid;   // B*T = 76800 = 9600*8 exactly

  if (lane < NEDGE) {
    s_src[wid][lane] = ei[g * (2 * NEDGE) + lane];
    s_dst[wid][lane] = ei[g * (2 * NEDGE) + NEDGE + lane];
  }
  float h0[2] = {0.f, 0.f};
  if (lane < NKP) {
    h0[0] = x[(g * NKP + lane) * 2 + 0];
    h0[1] = x[(g * NKP + lane) * 2 + 1];
  }
  __syncthreads();
  if (lane < NKP) {
    float deg = 1.f;                           // self loop
    for (int e = 0; e < NEDGE; ++e) deg += (s_dst[wid][e] == lane) ? 1.f : 0.f;
    s_dinv[wid][lane] = rsqrtf(deg);
  }
  __syncthreads();

  float h1[16], h2[16], h3[4];
  gcn_layer<2, 16>(s_xw[wid], s_dinv[wid], s_src[wid], s_dst[wid], lane, h0, W1, b1, h1);
  gcn_layer<16,16>(s_xw[wid], s_dinv[wid], s_src[wid], s_dst[wid], lane, h1, W2, b2, h2);
  gcn_layer<16, 4>(s_xw[wid], s_dinv[wid], s_src[wid], s_dst[wid], lane, h2, W3, b3, h3);

  if (lane < NKP) {
    for (int c = 0; c < 4; ++c) x_time[g * 68 + lane * 4 + c] = h3[c];
  }
}

// ---------------- Stage 2: implicit-GEMM conv / FC via f32 WMMA ----------------
// out[b, n, l] = sum_{c,kw} in[b, c, l*stride - pad + kw] * W[n, c, kw] + bias[n]
// M = B*Lout (tiles of 16), N = Ochan (tiles of 16), K = Cin*Kw (steps of 4).
// Input addressing: in[b*sb + c*sc + t*st]  (supports [B,T,C], [B,C,L], flat [B,K]).
// One wave per 16x16 output tile; V_WMMA_F32_16X16X4_F32 in fp32 (full precision
// vs the fp32 reference). 4x-unrolled K loop keeps the XDL pipe fed; weight
// stream prefetched (global_prefetch_b8).
__global__ void gemm_conv_wmma(const float* __restrict__ in,
                               const float* __restrict__ w,
                               const float* __restrict__ bias,
                               float* __restrict__ out,
                               int Lin, int Lout, int Ochan,
                               int Kw, int stride, int pad, int Ktot,
                               long sb, long sc, long st,
                               int w_kmajor)
{
  const int lane = threadIdx.x;          // blockDim = 32 (one wave, EXEC all-ones)
  const int rA   = lane & 15;
  const int hi   = lane >> 4;            // 0: K=k0,k0+1 ; 1: K=k0+2,k0+3
  const int mt   = blockIdx.x * 16;
  const int nt   = blockIdx.y * 16;

  // A row for this lane
  const int m    = mt + rA;
  const int bidx = m / Lout;
  const int l    = m - bidx * Lout;
  const int tbase = l * stride - pad;
  const long inb = (long)bidx * sb;
  // B column for this lane
  const int n    = nt + rA;

  v8f acc = {0.f,0.f,0.f,0.f,0.f,0.f,0.f,0.f};

#pragma unroll 4
  for (int k0 = 0; k0 < Ktot; k0 += 4) {
    const int kA = k0 + (hi << 1);
    // prefetch upcoming weight cachelines (L2-resident working set)
    if (w_kmajor) {
      __builtin_prefetch(&w[(long)(kA + 64) * Ochan + n], 0, 1);
    } else {
      __builtin_prefetch(&w[(long)n * Ktot + kA + 64], 0, 1);
    }
    // --- A fragment (im2col gather, zero-padded) ---
    v2f a;
    {
      int c0 = kA / Kw, kw0 = kA - c0 * Kw;
      int t0 = tbase + kw0;
      a.x = (t0 >= 0 && t0 < Lin) ? in[inb + (long)c0 * sc + (long)t0 * st] : 0.f;
      int kA1 = kA + 1;
      int c1 = kA1 / Kw, kw1 = kA1 - c1 * Kw;
      int t1 = tbase + kw1;
      a.y = (t1 >= 0 && t1 < Lin) ? in[inb + (long)c1 * sc + (long)t1 * st] : 0.f;
    }
    // --- B fragment (weights) ---
    v2f bf;
    if (w_kmajor) {
      bf.x = w[(long)kA * Ochan + n];
      bf.y = w[(long)(kA + 1) * Ochan + n];
    } else {
      bf.x = w[(long)n * Ktot + kA];
      bf.y = w[(long)n * Ktot + kA + 1];
    }
    acc = __builtin_amdgcn_wmma_f32_16x16x4_f32(false, a, false, bf,
                                                (short)0, acc, false, false);
  }

  // store D: VGPR i -> row (mt + i + 8*hi), col (nt + rA); fuse bias
  const float bv = bias[nt + rA];
  const int mrow0 = mt + (hi << 3);
#pragma unroll
  for (int i = 0; i < 8; ++i) {
    int mr = mrow0 + i;
    int bo = mr / Lout;
    int lo = mr - bo * Lout;
    out[(long)bo * Ochan * Lout + (long)(nt + rA) * Lout + lo] = acc[i] + bv;
  }
}

// ---------------- BatchNorm (training-mode batch stats) + ReLU ----------------
__global__ void bn_ncl_relu(float* __restrict__ data, const float* __restrict__ g,
                            const float* __restrict__ be, int Batch, int C, int L)
{
  const int c = blockIdx.x;
  const int tid = threadIdx.x;
  const int NL = Batch * L;
  float s = 0.f, q = 0.f;
  for (int i = tid; i < NL; i += BDIM) {
    int b = i / L, l = i - b * L;
    float v = data[(long)b * C * L + (long)c * L + l];
    s += v; q += v * v;
  }
  __shared__ float ss[BDIM], sq[BDIM];
  ss[tid] = s; sq[tid] = q;
  __syncthreads();
  for (int stp = BDIM / 2; stp > 0; stp >>= 1) {
    if (tid < stp) { ss[tid] += ss[tid + stp]; sq[tid] += sq[tid + stp]; }
    __syncthreads();
  }
  float m  = ss[0] / (float)NL;
  float vr = sq[0] / (float)NL - m * m;
  float scl = g[c] * rsqrtf(vr + EPSV);
  float sh  = be[c] - m * scl;
  for (int i = tid; i < NL; i += BDIM) {
    int b = i / L, l = i - b * L;
    long idx = (long)b * C * L + (long)c * L + l;
    data[idx] = fmaxf(data[idx] * scl + sh, 0.f);
  }
}

__global__ void bn_nc_relu(float* __restrict__ data, const float* __restrict__ g,
                           const float* __restrict__ be, int Batch, int C)
{
  const int c = blockIdx.x;
  const int b = threadIdx.x;   // blockDim == Batch == 256
  float v = data[(long)b * C + c];
  __shared__ float ss[BDIM], sq[BDIM];
  ss[b] = v; sq[b] = v * v;
  __syncthreads();
  for (int stp = BDIM / 2; stp > 0; stp >>= 1) {
    if (b < stp) { ss[b] += ss[b + stp]; sq[b] += sq[b + stp]; }
    __syncthreads();
  }
  float m  = ss[0] / (float)Batch;
  float vr = sq[0] / (float)Batch - m * m;
  data[(long)b * C + c] = fmaxf(g[c] * (v - m) * rsqrtf(vr + EPSV) + be[c], 0.f);
}

// ---------------- classification heads ----------------
__global__ void heads_kernel(const float* __restrict__ r,  // [256,16] (post BN+ReLU)
                             const float* __restrict__ Wi, const float* __restrict__ bi,
                             const float* __restrict__ Wa, const float* __restrict__ ba,
                             const float* __restrict__ Wc, const float* __restrict__ bc,
                             float* __restrict__ out)
{
  const int b = threadIdx.x;   // 256 threads
  float rv[16];
  for (int i = 0; i < 16; ++i) rv[i] = r[b * 16 + i];
  for (int j = 0; j < 3; ++j) {
    float s = bi[j];
    for (int i = 0; i < 16; ++i) s += rv[i] * Wi[i * 3 + j];
    out[b * 3 + j] = s;
  }
  for (int j = 0; j < 3; ++j) {
    float s = ba[j];
    for (int i = 0; i < 16; ++i) s += rv[i] * Wa[i * 3 + j];
    out[768 + b * 3 + j] = s;
  }
  for (int j = 0; j < 9; ++j) {
    float s = bc[j];
    for (int i = 0; i < 16; ++i) s += rv[i] * Wc[i * 9 + j];
    out[1536 + b * 9 + j] = s;
  }
}

// ---------------- launch ----------------
extern "C" void kernel_launch(void* const* d_in, const int* in_sizes, int n_in,
                              void* d_out, int out_size, void* d_ws, size_t ws_size,
                              hipStream_t stream) {
  (void)in_sizes; (void)n_in; (void)out_size; (void)ws_size;
  const float* x   = (const float*)d_in[0];
  const int*   ei  = (const int*)  d_in[1];
  // d_in[2] = edge_attr (unused by the reference)
  const float* W1  = (const float*)d_in[3];  const float* b1  = (const float*)d_in[4];
  const float* W2  = (const float*)d_in[5];  const float* b2  = (const float*)d_in[6];
  const float* W3  = (const float*)d_in[7];  const float* b3  = (const float*)d_in[8];
  const float* K1  = (const float*)d_in[9];  const float* cb1 = (const float*)d_in[10];
  const float* g1  = (const float*)d_in[11]; const float* be1 = (const float*)d_in[12];
  const float* K2  = (const float*)d_in[13]; const float* cb2 = (const float*)d_in[14];
  const float* g2  = (const float*)d_in[15]; const float* be2 = (const float*)d_in[16];
  const float* K3  = (const float*)d_in[17]; const float* cb3 = (const float*)d_in[18];
  const float* g3  = (const float*)d_in[19]; const float* be3 = (const float*)d_in[20];
  const float* Wf1 = (const float*)d_in[21]; const float* bf1 = (const float*)d_in[22];
  const float* gf1 = (const float*)d_in[23]; const float* bef1= (const float*)d_in[24];
  const float* Wf2 = (const float*)d_in[25]; const float* bf2 = (const float*)d_in[26];
  const float* gf2 = (const float*)d_in[27]; const float* bef2= (const float*)d_in[28];
  const float* Wi  = (const float*)d_in[29]; const float* bi  = (const float*)d_in[30];
  const float* Wa  = (const float*)d_in[31]; const float* ba  = (const float*)d_in[32];
  const float* Wc  = (const float*)d_in[33]; const float* bc  = (const float*)d_in[34];

  float* ws     = (float*)d_ws;
  float* x_time = ws;                     // [256*300, 68]  = 5,222,400
  float* y1     = x_time + 5222400;       // [256,256,100]  = 6,553,600
  float* y2     = y1     + 6553600;       // [256,128, 50]  = 1,638,400
  float* y3     = y2     + 1638400;       // [256, 64, 25]  =   409,600
  float* f1     = y3     + 409600;        // [256, 64]      =    16,384
  float* f2     = f1     + 16384;         // [256, 16]      =     4,096

  // 1) per-frame GCN: 76800 graphs, 8 per block
  gcn3_kernel<<<9600, 256, 0, stream>>>(x, ei, W1, b1, W2, b2, W3, b3, x_time);

  // 2) conv1: 68->256, k=7, s=3, p=3; in is [B,T,68] -> sb=20400, sc=1, st=68
  gemm_conv_wmma<<<dim3(1600, 16), 32, 0, stream>>>(x_time, K1, cb1, y1,
      300, 100, 256, 7, 3, 3, 476, 20400L, 1L, 68L, 0);
  bn_ncl_relu<<<256, BDIM, 0, stream>>>(y1, g1, be1, 256, 256, 100);

  // 3) conv2: 256->128, k=5, s=2, p=2; in [B,256,100] -> sb=25600, sc=100, st=1
  gemm_conv_wmma<<<dim3(800, 8), 32, 0, stream>>>(y1, K2, cb2, y2,
      100, 50, 128, 5, 2, 2, 1280, 25600L, 100L, 1L, 0);
  bn_ncl_relu<<<128, BDIM, 0, stream>>>(y2, g2, be2, 256, 128, 50);

  // 4) conv3: 128->64, k=5, s=2, p=2; in [B,128,50] -> sb=12800, sc=50, st=1
  gemm_conv_wmma<<<dim3(400, 4), 32, 0, stream>>>(y2, K3, cb3, y3,
      50, 25, 64, 5, 2, 2, 640, 12800L, 50L, 1L, 0);
  bn_ncl_relu<<<64, BDIM, 0, stream>>>(y3, g3, be3, 256, 64, 25);

  // 5) fc1: [256,1600] @ [1600,64] (k-major weights); flatten of y3 is contiguous
  gemm_conv_wmma<<<dim3(16, 4), 32, 0, stream>>>(y3, Wf1, bf1, f1,
      1, 1, 64, 1, 1, 0, 1600, 1600L, 1L, 0L, 1);
  bn_nc_relu<<<64, BDIM, 0, stream>>>(f1, gf1, bef1, 256, 64);

  // 6) fc2: [256,64] @ [64,16]
  gemm_conv_wmma<<<dim3(16, 1), 32, 0, stream>>>(f1, Wf2, bf2, f2,
      1, 1, 16, 1, 1, 0, 64, 64L, 1L, 0L, 1);
  bn_nc_relu<<<16, BDIM, 0, stream>>>(f2, gf2, bef2, 256, 16);  // ReLU fused (r = relu(y))

  // 7) heads -> d_out (y1|y2|y3 concatenated: 768 + 768 + 2304 floats)
  heads_kernel<<<1, 256, 0, stream>>>(f2, Wi, bi, Wa, ba, Wc, bc, (float*)d_out);
}